// MessagePassing_68848325755642
// MI455X (gfx1250) — compile-verified
//
#include <hip/hip_runtime.h>

#define D_FEAT 32

// ---------------------------------------------------------------------------
// Kernel 1: zero the output (harness poisons d_out with 0xAA before timing).
// float4 stores -> global_store_b128, fully coalesced.
// ---------------------------------------------------------------------------
__global__ __launch_bounds__(256) void mp_zero_kernel(float4* __restrict__ out, int n4) {
    int i = blockIdx.x * blockDim.x + threadIdx.x;
    if (i < n4) {
        out[i] = make_float4(0.f, 0.f, 0.f, 0.f);
    }
}

// ---------------------------------------------------------------------------
// Kernel 2: edge scatter-add.
//
// wave32 layout: lane = feature index (D_FEAT == 32 exactly).
// Each wave owns 32 consecutive edges:
//   - lanes cooperatively load 32 rows + 32 cols (2 coalesced b32 loads)
//   - per edge, broadcast (row, col) with v_readlane (constant lane -> SGPR),
//     making the gather/scatter base addresses wave-uniform: address math
//     runs on the (otherwise idle) SALU, loads use saddr+lane*4 form
//   - gather   : x[col*32 + lane]      -> one coalesced 128B global_load_b32
//   - scatter  : out[row*32 + lane] += -> one wave-wide global_atomic_add_f32
//
// x (12.8 MB) and out (12.8 MB) are L2-resident on MI455X (192 MB L2); the
// limiter is L2 load/atomic throughput, so minimize per-edge instruction
// overhead and keep 62.5k independent waves in flight to hide L2 latency.
// ---------------------------------------------------------------------------
__global__ __launch_bounds__(256) void mp_scatter_kernel(
    const float* __restrict__ x,
    const int*   __restrict__ rows,   // edge_index[0] : scatter targets
    const int*   __restrict__ cols,   // edge_index[1] : gather sources
    float*       __restrict__ out,
    int n_edges)
{
    const int lane = threadIdx.x & 31;
    const int wave = (blockIdx.x * (blockDim.x >> 5)) + (threadIdx.x >> 5);
    const long long base = (long long)wave * 32;
    if (base >= n_edges) return;

    const int remain = (int)(n_edges - base);
    const int count  = remain < 32 ? remain : 32;

    // Coalesced index loads: 32 edges' row/col per wave.
    int r = 0, c = 0;
    if (lane < count) {
        r = rows[base + lane];
        c = cols[base + lane];
    }

    if (count == 32) {
        // Fast path (all waves when E % 32 == 0): fully unrolled, constant
        // readlane -> SGPR broadcast, scalar address math, 32 loads + 32
        // atomics the scheduler can pipeline freely.
#pragma unroll
        for (int i = 0; i < 32; ++i) {
            const int ri = __builtin_amdgcn_readlane(r, i);  // SGPR, uniform
            const int ci = __builtin_amdgcn_readlane(c, i);  // SGPR, uniform
            const float v = x[ci * D_FEAT + lane];           // saddr + lane*4
            atomicAdd(out + (ri * D_FEAT + lane), v);        // global_atomic_add_f32
        }
    } else {
        // Generic tail (not hit for E = 2M, kept for correctness).
        for (int i = 0; i < count; ++i) {
            const int ri = __shfl(r, i);
            const int ci = __shfl(c, i);
            const float v = x[ci * D_FEAT + lane];
            atomicAdd(out + (ri * D_FEAT + lane), v);
        }
    }
}

// ---------------------------------------------------------------------------
// Launch wrapper.
// Inputs (setup_inputs order):
//   d_in[0] : x          float32 [N_NODES, 32]
//   d_in[1] : edge_index integer [2, N_EDGES]  (row-major: rows then cols)
// Output:
//   d_out   : float32 [N_NODES, 32]
// ---------------------------------------------------------------------------
extern "C" void kernel_launch(void* const* d_in, const int* in_sizes, int n_in,
                              void* d_out, int out_size, void* d_ws, size_t ws_size,
                              hipStream_t stream) {
    const float* x    = (const float*)d_in[0];
    const int*   eidx = (const int*)d_in[1];          // harness integer convention: int32
    const int    n_edges = in_sizes[1] / 2;
    const int*   rows = eidx;                         // edge_index[0]
    const int*   cols = eidx + n_edges;               // edge_index[1]
    float*       out  = (float*)d_out;

    // Zero-init output (out_size = N_NODES * 32, multiple of 4).
    const int n4 = out_size / 4;
    const int zblocks = (n4 + 255) / 256;
    mp_zero_kernel<<<zblocks, 256, 0, stream>>>((float4*)out, n4);

    // One wave per 32 edges; 8 waves per 256-thread block.
    const int n_waves  = (n_edges + 31) / 32;
    const int sblocks  = (n_waves + 7) / 8;
    mp_scatter_kernel<<<sblocks, 256, 0, stream>>>(x, rows, cols, out, n_edges);
}